// DifferentiableSelector_7705171329190
// MI455X (gfx1250) — compile-verified
//
#include <hip/hip_runtime.h>

#define T_DIM   32768
#define K_BUDGET 64.0f
#define CHUNK   4096
#define HALO    16
#define BPR     (T_DIM / CHUNK)   // 8 blocks per row

typedef float v2f __attribute__((ext_vector_type(2)));
typedef float v8f __attribute__((ext_vector_type(8)));
typedef int   v4i __attribute__((ext_vector_type(4)));
typedef __attribute__((address_space(1))) v4i* gvec_ptr;   // global int4*
typedef __attribute__((address_space(3))) v4i* lvec_ptr;   // LDS int4*

__device__ __forceinline__ float sigmoidf_(float x) {
  return 1.0f / (1.0f + expf(-x));
}

__device__ __forceinline__ float temp_inv(const float* lt) {
  float temp = fminf(fmaxf(expf(lt[0]), 0.1f), 10.0f);
  return 1.0f / temp;
}

// ---------------------------------------------------------------------------
// Kernel 1: per-row sum of sigmoid(scores/temp) -> scale = min(K/budget, 1)
// One block (1024 threads = 32 waves) per row; wave reduction via f32 WMMA.
// ---------------------------------------------------------------------------
__global__ __launch_bounds__(1024)
void rowscale_kernel(const float* __restrict__ scores,
                     const float* __restrict__ logtemp,
                     float* __restrict__ rowscale) {
  const int row = blockIdx.x;
  const int t   = threadIdx.x;
  const float invT = temp_inv(logtemp);
  const float4* __restrict__ src =
      (const float4*)(scores + (size_t)row * T_DIM);

  float p = 0.0f;
#pragma unroll
  for (int i = 0; i < 8; ++i) {
    float4 v = src[i * 1024 + t];
    p += sigmoidf_(v.x * invT) + sigmoidf_(v.y * invT) +
         sigmoidf_(v.z * invT) + sigmoidf_(v.w * invT);
  }

  // ---- wave32 reduction ----
  float wavetot;
#if __has_builtin(__builtin_amdgcn_wmma_f32_16x16x4_f32)
  {
    // A(16x4): lane m (0..15) supplies K=0,1 of row m; lanes 16..31 supply K=2,3.
    // Put partial in K=0/K=2 slot, zero in the other => rowsum_m = p_m + p_{m+16}.
    // B = all-ones (layout-independent) => D[m][n] = rowsum_m.
    v2f a; a.x = p;    a.y = 0.0f;
    v2f b; b.x = 1.0f; b.y = 1.0f;
    v8f c = {0.0f, 0.0f, 0.0f, 0.0f, 0.0f, 0.0f, 0.0f, 0.0f};
    v8f d = __builtin_amdgcn_wmma_f32_16x16x4_f32(
        false, a, false, b, (short)0, c, false, false);
    // lanes 0..15: sum_v d[v] = sum_{m=0..7} rowsum_m ; lanes 16..31: m=8..15
    float s = d[0] + d[1] + d[2] + d[3] + d[4] + d[5] + d[6] + d[7];
    wavetot = __shfl(s, 0, 32) + __shfl(s, 16, 32);
  }
#else
  {
    float s = p;
#pragma unroll
    for (int o = 16; o > 0; o >>= 1) s += __shfl_down(s, o, 32);
    wavetot = __shfl(s, 0, 32);
  }
#endif

  __shared__ float wsum[32];
  const int wave = t >> 5, lane = t & 31;
  if (lane == 0) wsum[wave] = wavetot;
  __syncthreads();
  if (t == 0) {
    float tot = 0.0f;
#pragma unroll
    for (int i = 0; i < 32; ++i) tot += wsum[i];
    tot = fmaxf(tot, 1e-6f);
    rowscale[row] = fminf(K_BUDGET / tot, 1.0f);
  }
}

// ---------------------------------------------------------------------------
// Kernel 2: y = scale*sigmoid(scores/temp); 4 refractory damping sweeps in
// LDS (chunk of 4096 + 16-element wraparound halo); zero column 0; write out.
// ---------------------------------------------------------------------------
#if __has_builtin(__builtin_amdgcn_global_load_async_to_lds_b128) && \
    __has_builtin(__builtin_amdgcn_s_wait_asynccnt)
#define USE_ASYNC_LDS 1
#endif

__global__ __launch_bounds__(256)
void damp_kernel(const float* __restrict__ scores,
                 const float* __restrict__ logtemp,
                 const float* __restrict__ rowscale,
                 float* __restrict__ out) {
  __shared__ __align__(16) float buf[CHUNK + HALO];

  const int row   = blockIdx.x / BPR;
  const int start = (blockIdx.x % BPR) * CHUNK;
  const int t     = threadIdx.x;
  const float invT = temp_inv(logtemp);
  const float sc   = rowscale[row];
  const float* __restrict__ rowp = scores + (size_t)row * T_DIM;

  // ---- stage raw scores (chunk + halo) into LDS ----
#ifdef USE_ASYNC_LDS
#pragma unroll
  for (int i = 0; i < 4; ++i) {
    const int off = i * 1024 + t * 4;
    __builtin_amdgcn_global_load_async_to_lds_b128(
        (gvec_ptr)(rowp + start + off),
        (lvec_ptr)(buf + off), 0, 0);
  }
  if (t < HALO / 4) {
    const int col = (start + CHUNK + t * 4) & (T_DIM - 1);  // row wraparound
    __builtin_amdgcn_global_load_async_to_lds_b128(
        (gvec_ptr)(rowp + col),
        (lvec_ptr)(buf + CHUNK + t * 4), 0, 0);
  }
  __builtin_amdgcn_s_wait_asynccnt(0);
  __syncthreads();
#else
#pragma unroll
  for (int i = 0; i < 4; ++i) {
    const int off = i * 1024 + t * 4;
    *(float4*)(buf + off) = *(const float4*)(rowp + start + off);
  }
  if (t < HALO) {
    const int col = (start + CHUNK + t) & (T_DIM - 1);
    buf[CHUNK + t] = rowp[col];
  }
  __syncthreads();
#endif

  // ---- convert in place: y0 = scale * sigmoid(score * invT) ----
  // stride-256 assignment: each lane owns distinct banks, conflict-free.
#pragma unroll
  for (int i = 0; i < 16; ++i) {
    const int p = t + 256 * i;
    buf[p] = sc * sigmoidf_(buf[p] * invT);
  }
  if (t < HALO) {
    const int p = CHUNK + t;
    buf[p] = sc * sigmoidf_(buf[p] * invT);
  }
  __syncthreads();

  // ---- 4 damping sweeps; halo updated too so later sweeps read valid data ----
#pragma unroll
  for (int d = 1; d <= 4; ++d) {
    float y[16], n[16];
#pragma unroll
    for (int i = 0; i < 16; ++i) {
      const int p = t + 256 * i;
      y[i] = buf[p];
      n[i] = buf[p + d];
    }
    float hy = 0.0f, hn = 0.0f;
    const int hp = CHUNK + t;
    const bool doHalo = (t < HALO) && (hp + d < CHUNK + HALO);
    if (doHalo) { hy = buf[hp]; hn = buf[hp + d]; }
    __syncthreads();
#pragma unroll
    for (int i = 0; i < 16; ++i) {
      const int p = t + 256 * i;
      buf[p] = y[i] * fminf(2.0f / (1.0f + y[i] + n[i]), 1.0f);
    }
    if (doHalo) buf[hp] = hy * fminf(2.0f / (1.0f + hy + hn), 1.0f);
    __syncthreads();
  }

  // ---- write result (zero global column 0) ----
  float4* __restrict__ orow = (float4*)(out + (size_t)row * T_DIM + start);
#pragma unroll
  for (int i = 0; i < 4; ++i) {
    const int off = i * 1024 + t * 4;
    float4 r = *(const float4*)(buf + off);
    if (start + off == 0) r.x = 0.0f;   // only i==0, t==0, start==0
    orow[i * 256 + t] = r;
  }
}

// ---------------------------------------------------------------------------
extern "C" void kernel_launch(void* const* d_in, const int* in_sizes, int n_in,
                              void* d_out, int out_size, void* d_ws, size_t ws_size,
                              hipStream_t stream) {
  const float* scores  = (const float*)d_in[0];
  const float* logtemp = (const float*)d_in[1];
  float* rowscale = (float*)d_ws;      // B floats of scratch
  float* out      = (float*)d_out;
  const int B = in_sizes[0] / T_DIM;   // 512

  rowscale_kernel<<<B, 1024, 0, stream>>>(scores, logtemp, rowscale);
  damp_kernel<<<B * BPR, 256, 0, stream>>>(scores, logtemp, rowscale, out);
}